// PFNLayer_59064390255321
// MI455X (gfx1250) — compile-verified
//
#include <hip/hip_runtime.h>

typedef __attribute__((ext_vector_type(2))) float v2f;
typedef __attribute__((ext_vector_type(8))) float v8f;

#define NGROUP 5
#define PTS    32
#define CIN    10
#define KPAD   12
#define COUT   64
#define BLK    256

__device__ __forceinline__ int group_id(int n) {
  return (n < 2) ? 0 : (n < 3) ? 1 : (n < 4) ? 2 : (n < 16) ? 3 : 4;
}

// One wave computes the full 32x64 voxel GEMM with V_WMMA_F32_16X16X4_F32.
// A: 16x4 f32 (2 VGPR: lanes 0-15 K={0,1}, lanes 16-31 K={2,3})
// B: 4x16 f32 (2 VGPR: reg0 rows K0/K0+1 by lane half, reg1 rows +2)
// C/D: 16x16 f32, 8 VGPR (lanes 0-15 M=r, lanes 16-31 M=r+8)
__device__ __forceinline__ void gemm_voxel(const float* __restrict__ inp,  // [32][10]
                                           const float* __restrict__ Wg,  // LDS [12][64]
                                           int lane, v8f acc[2][4]) {
  const v8f vzero = {};
#pragma unroll
  for (int mt = 0; mt < 2; ++mt)
#pragma unroll
    for (int nt = 0; nt < 4; ++nt) acc[mt][nt] = vzero;

  const int half = lane >> 4;
  const int lcol = lane & 15;
#pragma unroll
  for (int ks = 0; ks < 3; ++ks) {
    const int ka = ks * 4 + half * 2;  // A K-base for this lane (even)
    v2f a[2];
#pragma unroll
    for (int mt = 0; mt < 2; ++mt) {
      const int p = mt * 16 + lcol;  // A row (= point index)
      if (ka < CIN) {                // Cin even -> float2 all-valid or all-pad
        a[mt].x = inp[p * CIN + ka];
        a[mt].y = inp[p * CIN + ka + 1];
      } else {
        a[mt].x = 0.0f; a[mt].y = 0.0f;
      }
    }
    const int kb = ks * 4 + half;  // B row for reg0 (padded rows 10,11 are zero)
#pragma unroll
    for (int nt = 0; nt < 4; ++nt) {
      v2f b;
      b.x = Wg[kb * COUT + nt * 16 + lcol];
      b.y = Wg[(kb + 2) * COUT + nt * 16 + lcol];
#pragma unroll
      for (int mt = 0; mt < 2; ++mt)
        acc[mt][nt] = __builtin_amdgcn_wmma_f32_16x16x4_f32(
            false, a[mt], false, b, (short)0, acc[mt][nt], false, false);
    }
  }
}

__device__ __forceinline__ void load_W_lds(const float* __restrict__ W, float* sW) {
  for (int i = threadIdx.x; i < NGROUP * KPAD * COUT; i += blockDim.x) {
    int g = i / (KPAD * COUT);
    int k = (i / COUT) % KPAD;
    int c = i % COUT;
    sW[i] = (k < CIN) ? W[(g * CIN + k) * COUT + c] : 0.0f;
  }
}

__global__ void pfn_zero_kernel(float* __restrict__ p, int n) {
  int i = blockIdx.x * blockDim.x + threadIdx.x;
  if (i < n) p[i] = 0.0f;
}

__global__ void __launch_bounds__(BLK)
pfn_stats_kernel(const float* __restrict__ inp, const int* __restrict__ num,
                 const float* __restrict__ W, float* __restrict__ gsum,
                 float* __restrict__ gsq, float* __restrict__ gcnt, int N) {
  __shared__ float sW[NGROUP * KPAD * COUT];
  __shared__ float sSum[NGROUP * COUT];
  __shared__ float sSq[NGROUP * COUT];
  __shared__ float sCnt[NGROUP];
  load_W_lds(W, sW);
  for (int i = threadIdx.x; i < NGROUP * COUT; i += blockDim.x) {
    sSum[i] = 0.0f; sSq[i] = 0.0f;
  }
  if (threadIdx.x < NGROUP) sCnt[threadIdx.x] = 0.0f;
  __syncthreads();

  const int lane = threadIdx.x & 31;
  const int half = lane >> 4;
  const int lcol = lane & 15;
  const int wave = blockIdx.x * (blockDim.x >> 5) + (threadIdx.x >> 5);
  const int nw   = gridDim.x * (blockDim.x >> 5);

  for (int v = wave; v < N; v += nw) {   // wave-uniform: EXEC all-ones at WMMA
    const int g = group_id(num[v]);
    v8f acc[2][4];
    gemm_voxel(inp + (size_t)v * (PTS * CIN), sW + g * (KPAD * COUT), lane, acc);
#pragma unroll
    for (int nt = 0; nt < 4; ++nt) {
      float s = 0.0f, q = 0.0f;
#pragma unroll
      for (int mt = 0; mt < 2; ++mt)
#pragma unroll
        for (int r = 0; r < 8; ++r) {
          float x = acc[mt][nt][r];
          s += x; q += x * x;
        }
      // lanes L and L+16 hold the same output column -> pair-combine
      s += __shfl_xor(s, 16, 32);
      q += __shfl_xor(q, 16, 32);
      if (half == 0) {
        atomicAdd(&sSum[g * COUT + nt * 16 + lcol], s);
        atomicAdd(&sSq[g * COUT + nt * 16 + lcol], q);
      }
    }
    if (lane == 0) atomicAdd(&sCnt[g], (float)PTS);
  }
  __syncthreads();
  for (int i = threadIdx.x; i < NGROUP * COUT; i += blockDim.x) {
    atomicAdd(&gsum[i], sSum[i]);
    atomicAdd(&gsq[i], sSq[i]);
  }
  if (threadIdx.x < NGROUP) atomicAdd(&gcnt[threadIdx.x], sCnt[threadIdx.x]);
}

__global__ void pfn_scale_kernel(const float* __restrict__ gsum,
                                 const float* __restrict__ gsq,
                                 const float* __restrict__ gcnt,
                                 const float* __restrict__ gamma,
                                 const float* __restrict__ beta,
                                 float* __restrict__ scale,
                                 float* __restrict__ shift) {
  int i = threadIdx.x;
  if (i < NGROUP * COUT) {
    int g = i / COUT;
    float denom = fmaxf(gcnt[g], 1.0f);
    float mean = gsum[i] / denom;
    float var = gsq[i] / denom - mean * mean;
    float sc = gamma[i] * rsqrtf(var + 1e-3f);
    scale[i] = sc;
    shift[i] = beta[i] - mean * sc;
  }
}

__global__ void __launch_bounds__(BLK)
pfn_out_kernel(const float* __restrict__ inp, const int* __restrict__ num,
               const float* __restrict__ W, const float* __restrict__ scale,
               const float* __restrict__ shift, float* __restrict__ out, int N) {
  __shared__ float sW[NGROUP * KPAD * COUT];
  __shared__ float sScale[NGROUP * COUT];
  __shared__ float sShift[NGROUP * COUT];
  load_W_lds(W, sW);
  for (int i = threadIdx.x; i < NGROUP * COUT; i += blockDim.x) {
    sScale[i] = scale[i];
    sShift[i] = shift[i];
  }
  __syncthreads();

  const int lane = threadIdx.x & 31;
  const int half = lane >> 4;
  const int lcol = lane & 15;
  const int wave = blockIdx.x * (blockDim.x >> 5) + (threadIdx.x >> 5);
  const int nw   = gridDim.x * (blockDim.x >> 5);

  for (int v = wave; v < N; v += nw) {
    const int g = group_id(num[v]);
    v8f acc[2][4];
    gemm_voxel(inp + (size_t)v * (PTS * CIN), sW + g * (KPAD * COUT), lane, acc);
    float* __restrict__ ov = out + (size_t)v * (PTS * 2 * COUT);
#pragma unroll
    for (int nt = 0; nt < 4; ++nt) {
      const int col = nt * 16 + lcol;
      const float sc = sScale[g * COUT + col];
      const float sh = sShift[g * COUT + col];
      float m = 0.0f;  // ReLU outputs are >= 0
#pragma unroll
      for (int mt = 0; mt < 2; ++mt)
#pragma unroll
        for (int r = 0; r < 8; ++r) {
          float y = fmaxf(fmaf(acc[mt][nt][r], sc, sh), 0.0f);
          acc[mt][nt][r] = y;
          m = fmaxf(m, y);
        }
      m = fmaxf(m, __shfl_xor(m, 16, 32));  // column max over all 32 points
#pragma unroll
      for (int mt = 0; mt < 2; ++mt)
#pragma unroll
        for (int r = 0; r < 8; ++r) {
          const int p = mt * 16 + r + half * 8;
          ov[p * 128 + col] = acc[mt][nt][r];
          ov[p * 128 + 64 + col] = m;
        }
    }
  }
}

extern "C" void kernel_launch(void* const* d_in, const int* in_sizes, int n_in,
                              void* d_out, int out_size, void* d_ws, size_t ws_size,
                              hipStream_t stream) {
  (void)n_in; (void)out_size; (void)ws_size;
  const float* inp   = (const float*)d_in[0];
  const int*   num   = (const int*)d_in[1];
  const float* W     = (const float*)d_in[2];
  const float* gamma = (const float*)d_in[3];
  const float* beta  = (const float*)d_in[4];
  float* out = (float*)d_out;
  const int N = in_sizes[1];  // 40000 voxels

  float* ws    = (float*)d_ws;
  float* gsum  = ws;          // [5][64]
  float* gsq   = ws + 320;    // [5][64]
  float* gcnt  = ws + 640;    // [5] (+3 pad)
  float* scl   = ws + 648;    // [5][64]
  float* shf   = ws + 968;    // [5][64]

  pfn_zero_kernel<<<3, 256, 0, stream>>>(ws, 648);

  const int nblk = 640;  // 5120 waves, ~8 voxels/wave, W-LDS load amortized
  pfn_stats_kernel<<<nblk, BLK, 0, stream>>>(inp, num, W, gsum, gsq, gcnt, N);
  pfn_scale_kernel<<<1, NGROUP * COUT, 0, stream>>>(gsum, gsq, gcnt, gamma, beta, scl, shf);
  pfn_out_kernel<<<nblk, BLK, 0, stream>>>(inp, num, W, scl, shf, out, N);
}